// H_gcn_43147241455752
// MI455X (gfx1250) — compile-verified
//
#include <hip/hip_runtime.h>
#include <math.h>

typedef __attribute__((ext_vector_type(2))) float v2f;
typedef __attribute__((ext_vector_type(8))) float v8f;

#define LRELU_SLOPE 0.2f

__device__ __forceinline__ float lrelu(float v) { return v > 0.0f ? v : LRELU_SLOPE * v; }

// Monotone order-preserving float <-> uint mapping for atomicMax on floats.
__device__ __forceinline__ unsigned mapf(float f) {
  unsigned u = __float_as_uint(f);
  return (u & 0x80000000u) ? ~u : (u | 0x80000000u);
}
__device__ __forceinline__ float unmapf(unsigned u) {
  return __uint_as_float((u & 0x80000000u) ? (u & 0x7FFFFFFFu) : ~u);
}

__device__ __forceinline__ float waveReduceAdd(float v) {
#pragma unroll
  for (int off = 16; off > 0; off >>= 1) v += __shfl_xor(v, off, 32);
  return v;
}

// ---------------- zero init ----------------
__global__ void k_zero(unsigned* __restrict__ p, long long n) {
  long long i = (long long)blockIdx.x * blockDim.x + threadIdx.x;
  long long stride = (long long)gridDim.x * blockDim.x;
  for (; i < n; i += stride) p[i] = 0u;
}

// ---------------- SpMM: dst[row[e]] += w[e] * src[col[e]]  (one wave per edge) ----------------
__global__ void k_spmm(const long long* __restrict__ row, const long long* __restrict__ col,
                       const float* __restrict__ w, const float* __restrict__ src, int sstride,
                       float* __restrict__ dst, int nedges) {
  long long gid = (long long)blockIdx.x * blockDim.x + threadIdx.x;
  int lane = threadIdx.x & 31;
  long long e = gid >> 5;
  long long wstride = ((long long)gridDim.x * blockDim.x) >> 5;
  for (; e < nedges; e += wstride) {
    int r = (int)row[e], c = (int)col[e];
    float wt = w[e];
    float4 v = *(const float4*)(src + (long long)c * sstride + (lane << 2));
    float* dp = dst + (long long)r * 128 + (lane << 2);
    atomicAdd(dp + 0, wt * v.x);
    atomicAdd(dp + 1, wt * v.y);
    atomicAdd(dp + 2, wt * v.z);
    atomicAdd(dp + 3, wt * v.w);
  }
}

// ---------------- tr accumulation: sum(y * m) ----------------
__global__ void k_dot_tr(const float* __restrict__ y128, const float* __restrict__ m,
                         float* __restrict__ acc, int n) {
  long long total = (long long)n * 128;
  long long i = (long long)blockIdx.x * blockDim.x + threadIdx.x;
  long long stride = (long long)gridDim.x * blockDim.x;
  float s = 0.0f;
  for (; i < total; i += stride) {
    long long node = i >> 7;
    int c = (int)(i & 127);
    s += y128[node * 256 + c] * m[i];
  }
  s = waveReduceAdd(s);
  if ((threadIdx.x & 31) == 0) atomicAdd(acc, s);
}

// ---------------- h = x @ W via V_WMMA_F32_16X16X4_F32 ----------------
// Block = 256 threads = 8 waves = one 16-row stripe of x, each wave one 16-col tile of W.
// x stripe staged in LDS, padded stride 132 (conflict-free A reads).
#define SX_STRIDE 132
__global__ __launch_bounds__(256) void k_gemm_wmma(const float* __restrict__ xf,
                                                   const float* __restrict__ Wm,
                                                   float* __restrict__ h) {
  __shared__ float sX[16 * SX_STRIDE];
  const int tid = threadIdx.x;
  const int tile_m = blockIdx.x;
  // cooperative load of 16x128 x-tile (row stride 256 floats in x_full)
  for (int i4 = tid; i4 < 512; i4 += 256) {
    int r = i4 >> 5, c4 = (i4 & 31) << 2;
    *(float4*)&sX[r * SX_STRIDE + c4] =
        *(const float4*)(xf + (long long)(tile_m * 16 + r) * 256 + c4);
  }
  __syncthreads();

  const int wave = tid >> 5;
  const int lane = tid & 31;
  const int m = lane & 15;
  const int koff = (lane >> 4) << 1;       // lanes 0-15 -> K {k,k+1}; 16-31 -> {k+2,k+3}
  const int ncol = (wave << 4) + m;        // column within the 128 output cols

  v8f acc = {};
#pragma unroll 4
  for (int k = 0; k < 128; k += 4) {
    v2f a = *(const v2f*)&sX[m * SX_STRIDE + k + koff];   // ds_load_b64
    v2f b;
    b.x = Wm[(k + koff) * 128 + ncol];
    b.y = Wm[(k + koff + 1) * 128 + ncol];
    acc = __builtin_amdgcn_wmma_f32_16x16x4_f32(false, a, false, b,
                                                (short)0, acc, false, false);
  }
  // D layout: VGPR i -> M=i (lanes 0-15) / M=8+i (lanes 16-31), N = lane&15
  const int mbase = tile_m * 16 + ((lane >> 4) << 3);
#pragma unroll
  for (int i = 0; i < 8; ++i) {
    h[(long long)(mbase + i) * 128 + ncol] = acc[i];
  }
}

// scalar fallback for a row remainder (not used when N % 16 == 0)
__global__ void k_gemm_tail(const float* __restrict__ xf, const float* __restrict__ Wm,
                            float* __restrict__ h, int startRow, int n) {
  int total = (n - startRow) * 128;
  int idx = blockIdx.x * blockDim.x + threadIdx.x;
  if (idx >= total) return;
  int r = startRow + (idx >> 7), c = idx & 127;
  float s = 0.0f;
  for (int k = 0; k < 128; ++k) s += xf[(long long)r * 256 + k] * Wm[k * 128 + c];
  h[(long long)r * 128 + c] = s;
}

// ---------------- a_s = h @ a_src, a_d = h @ a_dst (wave per node) ----------------
__global__ void k_asd(const float* __restrict__ h, const float* __restrict__ asrc,
                      const float* __restrict__ adst, float* __restrict__ as_,
                      float* __restrict__ ad_, int n) {
  long long gid = (long long)blockIdx.x * blockDim.x + threadIdx.x;
  int lane = threadIdx.x & 31;
  long long node = gid >> 5;
  long long wstride = ((long long)gridDim.x * blockDim.x) >> 5;
  float4 sa = *(const float4*)(asrc + (lane << 2));
  float4 da = *(const float4*)(adst + (lane << 2));
  for (; node < n; node += wstride) {
    float4 hv = *(const float4*)(h + node * 128 + (lane << 2));
    float s = hv.x * sa.x + hv.y * sa.y + hv.z * sa.z + hv.w * sa.w;
    float d = hv.x * da.x + hv.y * da.y + hv.z * da.z + hv.w * da.w;
    s = waveReduceAdd(s);
    d = waveReduceAdd(d);
    if (lane == 0) { as_[node] = s; ad_[node] = d; }
  }
}

// ---------------- edge softmax: pass 1 (segment max over dst) ----------------
__global__ void k_emax(const long long* __restrict__ row, const long long* __restrict__ col,
                       const float* __restrict__ as_, const float* __restrict__ ad_,
                       unsigned* __restrict__ emaxu, int E, int n) {
  long long total = (long long)E + n;
  long long t = (long long)blockIdx.x * blockDim.x + threadIdx.x;
  long long stride = (long long)gridDim.x * blockDim.x;
  for (; t < total; t += stride) {
    int s, d;
    if (t < E) { s = (int)row[t]; d = (int)col[t]; } else { s = d = (int)(t - E); }
    float v = lrelu(as_[s] + ad_[d]);
    atomicMax(emaxu + d, mapf(v));
  }
}

// ---------------- edge softmax: pass 2 (denominator) ----------------
__global__ void k_denom(const long long* __restrict__ row, const long long* __restrict__ col,
                        const float* __restrict__ as_, const float* __restrict__ ad_,
                        const unsigned* __restrict__ emaxu, float* __restrict__ denom,
                        int E, int n) {
  long long total = (long long)E + n;
  long long t = (long long)blockIdx.x * blockDim.x + threadIdx.x;
  long long stride = (long long)gridDim.x * blockDim.x;
  for (; t < total; t += stride) {
    int s, d;
    if (t < E) { s = (int)row[t]; d = (int)col[t]; } else { s = d = (int)(t - E); }
    float v = lrelu(as_[s] + ad_[d]);
    atomicAdd(denom + d, expf(v - unmapf(emaxu[d])));
  }
}

// ---------------- U[dst] += alpha * h[src]  (one wave per extended edge) ----------------
__global__ void k_scatter(const long long* __restrict__ row, const long long* __restrict__ col,
                          const float* __restrict__ as_, const float* __restrict__ ad_,
                          const unsigned* __restrict__ emaxu, const float* __restrict__ denom,
                          const float* __restrict__ h, float* __restrict__ U, int E, int n) {
  long long total = (long long)E + n;
  long long gid = (long long)blockIdx.x * blockDim.x + threadIdx.x;
  int lane = threadIdx.x & 31;
  long long t = gid >> 5;
  long long wstride = ((long long)gridDim.x * blockDim.x) >> 5;
  for (; t < total; t += wstride) {
    int s, d;
    if (t < E) { s = (int)row[t]; d = (int)col[t]; } else { s = d = (int)(t - E); }
    float v = lrelu(as_[s] + ad_[d]);
    float alpha = expf(v - unmapf(emaxu[d])) / (denom[d] + 1e-16f);
    float4 hv = *(const float4*)(h + (long long)s * 128 + (lane << 2));
    float* up = U + (long long)d * 128 + (lane << 2);
    atomicAdd(up + 0, alpha * hv.x);
    atomicAdd(up + 1, alpha * hv.y);
    atomicAdd(up + 2, alpha * hv.z);
    atomicAdd(up + 3, alpha * hv.w);
  }
}

// ---------------- sum of squares of (U + bias) ----------------
__global__ void k_sumsq(const float* __restrict__ U, const float* __restrict__ bias,
                        float* __restrict__ acc, int n) {
  long long total = (long long)n * 128;
  long long i = (long long)blockIdx.x * blockDim.x + threadIdx.x;
  long long stride = (long long)gridDim.x * blockDim.x;
  float s = 0.0f;
  for (; i < total; i += stride) {
    float v = U[i] + bias[(int)(i & 127)];
    s += v * v;
  }
  s = waveReduceAdd(s);
  if ((threadIdx.x & 31) == 0) atomicAdd(acc, s);
}

__global__ void k_final(const float* __restrict__ scal, float* __restrict__ out, int n) {
  out[0] = scal[0] / (float)n + sqrtf(scal[1]);
}

extern "C" void kernel_launch(void* const* d_in, const int* in_sizes, int n_in,
                              void* d_out, int out_size, void* d_ws, size_t ws_size,
                              hipStream_t stream) {
  const float* x_full   = (const float*)d_in[0];       // (N, 256)
  const long long* ei   = (const long long*)d_in[1];   // (2, E) int64
  const float* ew       = (const float*)d_in[2];       // (E,)
  const float* Wm       = (const float*)d_in[3];       // (128,128)
  const float* a_src    = (const float*)d_in[4];       // (128,)
  const float* a_dst    = (const float*)d_in[5];       // (128,)
  const float* bias     = (const float*)d_in[6];       // (128,)

  const int N = in_sizes[0] / 256;
  const int E = in_sizes[1] / 2;
  const long long* row = ei;
  const long long* col = ei + E;
  const float* y128 = x_full + 128;  // y = x_full[:, 128:], row stride 256

  // workspace layout (floats): A[N*128], B[N*128], a_s[N], a_d[N], emax[N], denom[N], scal[2]
  float* ws = (float*)d_ws;
  const long long nd = (long long)N * 128;
  float* A       = ws;             // out1, then h
  float* B       = A + nd;         // out2, then U
  float* as_     = B + nd;
  float* ad_     = as_ + N;
  unsigned* emaxu = (unsigned*)(ad_ + N);
  float* denom   = (float*)(emaxu + N);
  float* scal    = denom + N;      // [0]=tr accumulator, [1]=sumsq

  const int TB = 256;
  const unsigned gDense   = (unsigned)((nd + TB - 1) / TB);
  const unsigned gEdgeW   = (unsigned)(((long long)E * 32 + TB - 1) / TB);
  const unsigned gExt     = (unsigned)(((long long)(E + N) + TB - 1) / TB);
  const unsigned gExtW    = (unsigned)(((long long)(E + N) * 32 + TB - 1) / TB);
  const unsigned gNodeW   = (unsigned)(((long long)N * 32 + TB - 1) / TB);

  // ---- phase 1: out2 = A(A(y)), tr = sum(y*out2)/N ----
  k_zero<<<gDense, TB, 0, stream>>>((unsigned*)A, nd);
  k_spmm<<<gEdgeW, TB, 0, stream>>>(row, col, ew, y128, 256, A, E);
  k_zero<<<gDense, TB, 0, stream>>>((unsigned*)B, nd);
  k_spmm<<<gEdgeW, TB, 0, stream>>>(row, col, ew, A, 128, B, E);
  k_zero<<<1, 64, 0, stream>>>((unsigned*)scal, 2);
  k_dot_tr<<<gDense, TB, 0, stream>>>(y128, B, &scal[0], N);

  // ---- phase 2: h = x @ W (WMMA f32), attention scalars ----
  const int ntiles = N / 16;
  if (ntiles > 0) k_gemm_wmma<<<ntiles, TB, 0, stream>>>(x_full, Wm, A);
  const int rem = N % 16;
  if (rem > 0) {
    unsigned g = (unsigned)((rem * 128 + TB - 1) / TB);
    k_gemm_tail<<<g, TB, 0, stream>>>(x_full, Wm, A, ntiles * 16, N);
  }
  k_asd<<<gNodeW, TB, 0, stream>>>(A, a_src, a_dst, as_, ad_, N);

  // ---- phase 3: edge softmax + aggregation ----
  k_zero<<<(unsigned)((2LL * N + TB - 1) / TB), TB, 0, stream>>>(emaxu, 2LL * N); // emax + denom
  k_emax<<<gExt, TB, 0, stream>>>(row, col, as_, ad_, emaxu, E, N);
  k_denom<<<gExt, TB, 0, stream>>>(row, col, as_, ad_, emaxu, denom, E, N);
  k_zero<<<gDense, TB, 0, stream>>>((unsigned*)B, nd);   // U
  k_scatter<<<gExtW, TB, 0, stream>>>(row, col, as_, ad_, emaxu, denom, A, B, E, N);

  // ---- phase 4: scalar result ----
  k_sumsq<<<gDense, TB, 0, stream>>>(B, bias, &scal[1], N);
  k_final<<<1, 1, 0, stream>>>(scal, (float*)d_out, N);

  (void)n_in; (void)out_size; (void)ws_size;
}